// Ours_23132693856312
// MI455X (gfx1250) — compile-verified
//
#include <hip/hip_runtime.h>

typedef float v2f __attribute__((ext_vector_type(2)));
typedef float v8f __attribute__((ext_vector_type(8)));

#define HIDF  128
#define CATW  640
#define BNEPS 1e-5f
#define KCHUNK 128
#define LDSP  130   // padded LDS row stride (dwords): even (8B align) + bank spread

__device__ __forceinline__ v8f wmma4(v2f a, v2f b, v8f c) {
#if defined(__gfx1250__)
  // D = A(16x4,f32) * B(4x16,f32) + C(16x16,f32)
  return __builtin_amdgcn_wmma_f32_16x16x4_f32(false, a, false, b, (short)0, c, false, false);
#else
  (void)a; (void)b; return c;
#endif
}

// ---------------- utility / graph kernels ----------------

__global__ void k_zero(float* __restrict__ p, long n) {
  long i = (long)blockIdx.x * blockDim.x + threadIdx.x;
  long s = (long)gridDim.x * blockDim.x;
  for (; i < n; i += s) p[i] = 0.0f;
}

__global__ void k_deg(const int* __restrict__ col, float* __restrict__ deg, int E) {
  int i = blockIdx.x * blockDim.x + threadIdx.x;
  int s = gridDim.x * blockDim.x;
  for (; i < E; i += s) atomicAdd(&deg[col[i]], 1.0f);
}

__global__ void k_dis(float* __restrict__ deg, int n) {
  int i = blockIdx.x * blockDim.x + threadIdx.x;
  if (i < n) { float d = deg[i]; deg[i] = (d > 0.0f) ? rsqrtf(d) : 0.0f; }
}

__global__ void k_ew(const int* __restrict__ row, const int* __restrict__ col,
                     const float* __restrict__ dis, float* __restrict__ ew, int E) {
  int i = blockIdx.x * blockDim.x + threadIdx.x;
  int s = gridDim.x * blockDim.x;
  for (; i < E; i += s) ew[i] = dis[col[i]] * dis[row[i]];
}

// scatter: dst[col[e], :] += ew[e] * src[row[e], :]   (one wave per edge)
__global__ void k_prop(const int* __restrict__ rows, const int* __restrict__ cols,
                       const float* __restrict__ ew,
                       const float* __restrict__ src, int lds_,
                       float* __restrict__ dst, int ldd, int E) {
  int wave = threadIdx.x >> 5, lane = threadIdx.x & 31;
  int e = blockIdx.x * 8 + wave;                 // blockDim.x == 256 -> 8 waves
  if (e >= E) return;
  int r = rows[e], c = cols[e];
  float w = ew[e];
  const float4 v = *(const float4*)(src + (long)r * lds_ + lane * 4);
  float* d = dst + (long)c * ldd + lane * 4;
  atomicAdd(d + 0, w * v.x);
  atomicAdd(d + 1, w * v.y);
  atomicAdd(d + 2, w * v.z);
  atomicAdd(d + 3, w * v.w);
}

// ---------------- LDS-staged WMMA f32 GEMM ----------------
// C[M x NC] = A[M x K] @ W[K x NC] (+bias)(+R)
// block = 128 threads (4 waves); macro-tile = 64 rows x 64 cols.
// Wave w owns rows [m0+16w, m0+16w+16) and all four 16-col tiles:
// per K-step 1 A-frag + 4 B-frags feed 4 wmma (A reuse -> 1.25 ds loads/wmma).
// Requires K % 128 == 0 (all uses: 128, 640).  grid = (ceil(M/64), ceil(NC/64))
__launch_bounds__(128)
__global__ void k_gemm(const float* __restrict__ A, int lda,
                       const float* __restrict__ W,
                       const float* __restrict__ bias,
                       const float* __restrict__ R, int ldr,
                       float* __restrict__ C, int ldc,
                       int M, int K, int NC) {
  __shared__ float sB[64 * LDSP];   // sB[c * LDSP + kk] = W[k0+kk][cb+c]
  __shared__ float sA[64 * LDSP];   // sA[m * LDSP + kk] = A[m0+m][k0+kk]

  const int tid  = threadIdx.x;
  const int lane = tid & 31;
  const int wave = tid >> 5;
  const int m0 = blockIdx.x << 6;
  const int cb = blockIdx.y << 6;                 // block col base
  const int kg = 2 * (lane >> 4);                 // K-pair base within fragment
  const bool ncFull = (cb + 64 <= NC);

  // A staging coords: 2 threads per row, 64 floats each
  const int srow = tid >> 1;
  const int scol = (tid & 1) << 6;
  const int arow = m0 + srow;
  const int arowC = (arow < M) ? arow : (M - 1);
  const float* aStage = A + (long)arowC * lda + scol;

  // B staging coords: one W row per thread, 64 cols (transposed into LDS)
  const float* wStage = W + (long)tid * NC + cb;
  const int cClamp = (NC - 1) - cb;               // >= 0 (cb < NC always)

  const float* aFrag = sA + (wave * 16 + (lane & 15)) * LDSP + kg;
  const float* bFrag = sB + (lane & 15) * LDSP + kg;   // + j*16*LDSP per col tile

  v8f acc0 = {}, acc1 = {}, acc2 = {}, acc3 = {};

  for (int k0 = 0; k0 < K; k0 += KCHUNK) {
    __syncthreads();
    // ---- stage A tile: 64 x 128, coalesced float4 loads, float2 LDS stores ----
    {
      const float* ap = aStage + k0;
      float* dst = sA + srow * LDSP + scol;
#pragma unroll
      for (int j = 0; j < 16; ++j) {
        float4 v = *(const float4*)(ap + 4 * j);
        v2f lo; lo.x = v.x; lo.y = v.y;
        v2f hi; hi.x = v.z; hi.y = v.w;
        *(v2f*)(dst + 4 * j) = lo;
        *(v2f*)(dst + 4 * j + 2) = hi;
      }
    }
    // ---- stage B tile (transposed): W[k0+tid][cb..cb+63] -> sB[c][tid] ----
    {
      const float* wr = wStage + (long)k0 * NC;
      float* dst = sB + tid;
      if (ncFull) {
#pragma unroll
        for (int j = 0; j < 16; ++j) {
          float4 v = *(const float4*)(wr + 4 * j);
          dst[(4 * j + 0) * LDSP] = v.x;
          dst[(4 * j + 1) * LDSP] = v.y;
          dst[(4 * j + 2) * LDSP] = v.z;
          dst[(4 * j + 3) * LDSP] = v.w;
        }
      } else {
        // clamped address + value mask: unconditional loads, no exec divergence
        for (int c = 0; c < 64; ++c) {
          int ci = (c <= cClamp) ? c : cClamp;
          float msk = (c <= cClamp) ? 1.0f : 0.0f;
          dst[c * LDSP] = wr[ci] * msk;
        }
      }
    }
    __syncthreads();
    // ---- compute: per K-step, 1 A-frag + 4 B-frags -> 4 wmma ----
#pragma unroll 4
    for (int kk = 0; kk < KCHUNK; kk += 4) {
      v2f a  = *(const v2f*)(aFrag + kk);
      v2f b0 = *(const v2f*)(bFrag + kk);
      v2f b1 = *(const v2f*)(bFrag + 16 * LDSP + kk);
      v2f b2 = *(const v2f*)(bFrag + 32 * LDSP + kk);
      v2f b3 = *(const v2f*)(bFrag + 48 * LDSP + kk);
      acc0 = wmma4(a, b0, acc0);
      acc1 = wmma4(a, b1, acc1);
      acc2 = wmma4(a, b2, acc2);
      acc3 = wmma4(a, b3, acc3);
    }
  }

  // ---- epilogue: 4 col tiles ----
  const int rw = m0 + (wave << 4);
  const int cg = lane >> 4;
  v8f accs[4] = {acc0, acc1, acc2, acc3};
#pragma unroll
  for (int j = 0; j < 4; ++j) {
    int col = cb + j * 16 + (lane & 15);
    if (col < NC) {
      float badd = bias ? bias[col] : 0.0f;
#pragma unroll
      for (int r = 0; r < 8; ++r) {
        int row = rw + r + 8 * cg;
        if (row < M) {
          float v = accs[j][r] + badd;
          if (R) v += R[(long)row * ldr + col];
          C[(long)row * ldc + col] = v;
        }
      }
    }
  }
}

// ---------------- kvs = kn^T @ h  (128x128), partials over N slices -----------------
// grid = (64 tiles, NSLICES), block = 32 (one wave). Requires n % 4 == 0.
__launch_bounds__(32)
__global__ void k_kvs(const float* __restrict__ KN, const float* __restrict__ H,
                      float* __restrict__ kvs, int n, int per) {
  const int lane = threadIdx.x & 31;
  const int mt = blockIdx.x & 7;
  const int dt = blockIdx.x >> 3;
  const int kg = 2 * (lane >> 4);
  const int mcol = mt * 16 + (lane & 15);
  const int dcol = dt * 16 + (lane & 15);
  const int n0 = blockIdx.y * per;
  int n1 = n0 + per; if (n1 > n) n1 = n;        // n % 4 == 0 -> all rows in bounds

  const float* kp = KN + (long)(n0 + kg) * HIDF + mcol;
  const float* hp = H  + (long)(n0 + kg) * HIDF + dcol;

  v8f acc = {};
  for (int nn = n0; nn < n1; nn += 4) {
    v2f a; a.x = kp[0]; a.y = kp[HIDF];
    v2f b; b.x = hp[0]; b.y = hp[HIDF];
    kp += 4 * HIDF; hp += 4 * HIDF;
    acc = wmma4(a, b, acc);
  }
  const int cg = lane >> 4;
#pragma unroll
  for (int r = 0; r < 8; ++r)
    atomicAdd(&kvs[(mt * 16 + r + 8 * cg) * HIDF + dcol], acc[r]);
}

// ---------------- row normalize (wave per row); optionally accumulate column sum ------
__global__ void k_rownorm(float* __restrict__ Q, int n, float* __restrict__ colsum) {
  int wave = threadIdx.x >> 5, lane = threadIdx.x & 31;
  int r = blockIdx.x * 4 + wave;                 // blockDim.x == 128
  if (r >= n) return;
  float4 v = *(const float4*)(Q + (long)r * HIDF + lane * 4);
  float s = v.x * v.x + v.y * v.y + v.z * v.z + v.w * v.w;
#pragma unroll
  for (int m = 16; m >= 1; m >>= 1) s += __shfl_xor(s, m, 32);
  float inv = rsqrtf(s);
  v.x *= inv; v.y *= inv; v.z *= inv; v.w *= inv;
  *(float4*)(Q + (long)r * HIDF + lane * 4) = v;
  if (colsum) {
    atomicAdd(&colsum[lane * 4 + 0], v.x);
    atomicAdd(&colsum[lane * 4 + 1], v.y);
    atomicAdd(&colsum[lane * 4 + 2], v.z);
    atomicAdd(&colsum[lane * 4 + 3], v.w);
  }
}

// column sums (and optional sum of squares) over [n x 128]; block = 128 threads
__global__ void k_colsum(const float* __restrict__ A, int ld, int n,
                         float* __restrict__ sum, float* __restrict__ sumsq) {
  int c = threadIdx.x;
  float s = 0.0f, s2 = 0.0f;
  for (int r = blockIdx.x; r < n; r += gridDim.x) {
    float v = A[(long)r * ld + c];
    s += v; s2 += v * v;
  }
  atomicAdd(&sum[c], s);
  if (sumsq) atomicAdd(&sumsq[c], s2);
}

__global__ void k_bnfin(const float* __restrict__ sum, const float* __restrict__ sumsq,
                        float* __restrict__ mean, float* __restrict__ rstd, float n) {
  int c = threadIdx.x;
  float m = sum[c] / n;
  float v = sumsq[c] / n - m * m;
  mean[c] = m;
  rstd[c] = rsqrtf(v + BNEPS);
}

__global__ void k_bnrelu(const float* __restrict__ z, const float* __restrict__ mean,
                         const float* __restrict__ rstd, const float* __restrict__ g,
                         const float* __restrict__ b, float* __restrict__ h, long total) {
  long i = (long)blockIdx.x * blockDim.x + threadIdx.x;
  long s = (long)gridDim.x * blockDim.x;
  for (; i < total; i += s) {
    int c = (int)(i & (HIDF - 1));
    h[i] = fmaxf(0.0f, g[c] * (z[i] - mean[c]) * rstd[c] + b[c]);
  }
}

// x1 = num / den;  den = qn . ksum + N ; num already resident in cat[:, 0:128]
__global__ void k_x1div(float* __restrict__ cat, const float* __restrict__ QN,
                        const float* __restrict__ ksum, int n) {
  int wave = threadIdx.x >> 5, lane = threadIdx.x & 31;
  int r = blockIdx.x * 4 + wave;
  if (r >= n) return;
  const float4 q = *(const float4*)(QN + (long)r * HIDF + lane * 4);
  const float4 ks = *(const float4*)(ksum + lane * 4);
  float s = q.x * ks.x + q.y * ks.y + q.z * ks.z + q.w * ks.w;
#pragma unroll
  for (int m = 16; m >= 1; m >>= 1) s += __shfl_xor(s, m, 32);
  float inv = 1.0f / (s + (float)n);
  float4 v = *(const float4*)(cat + (long)r * CATW + lane * 4);
  v.x *= inv; v.y *= inv; v.z *= inv; v.w *= inv;
  *(float4*)(cat + (long)r * CATW + lane * 4) = v;
}

// cat[:, 128:256] = h
__global__ void k_copy_h(const float* __restrict__ h, float* __restrict__ cat, long total) {
  long i = (long)blockIdx.x * blockDim.x + threadIdx.x;
  long s = (long)gridDim.x * blockDim.x;
  for (; i < total; i += s) {
    long r = i >> 7;
    int c = (int)(i & (HIDF - 1));
    cat[r * CATW + HIDF + c] = h[i];
  }
}

// ---------------- host driver ----------------

static void launch_gemm(const float* A, int lda, const float* W, const float* bias,
                        const float* R, int ldr, float* C, int ldc,
                        int M, int K, int NC, hipStream_t stream) {
  dim3 grid((M + 63) / 64, (NC + 63) / 64);
  k_gemm<<<grid, 128, 0, stream>>>(A, lda, W, bias, R, ldr, C, ldc, M, K, NC);
}

extern "C" void kernel_launch(void* const* d_in, const int* in_sizes, int n_in,
                              void* d_out, int out_size, void* d_ws, size_t ws_size,
                              hipStream_t stream) {
  (void)n_in; (void)out_size; (void)ws_size;
  const float* x     = (const float*)d_in[0];
  const int*   ei    = (const int*)d_in[1];
  const float* W_in  = (const float*)d_in[2];
  const float* b_in  = (const float*)d_in[3];
  const float* Wq    = (const float*)d_in[4];
  const float* bq    = (const float*)d_in[5];
  const float* Wk    = (const float*)d_in[6];
  const float* bk    = (const float*)d_in[7];
  const float* Wfc   = (const float*)d_in[8];
  const float* bfc   = (const float*)d_in[9];
  const float* W_out = (const float*)d_in[10];
  const float* b_out = (const float*)d_in[11];
  const float* bn_g  = (const float*)d_in[12];
  const float* bn_b  = (const float*)d_in[13];

  const int N = in_sizes[0] / HIDF;   // 50000
  const int E = in_sizes[1] / 2;      // 800000
  const int OUTC = in_sizes[11];      // 40

  const int* erow = ei;
  const int* ecol = ei + E;

  // workspace carve-out (256B aligned)
  char* wsB = (char*)d_ws;
  size_t off = 0;
  auto alloc = [&](size_t nf) -> float* {
    float* p = (float*)(wsB + off);
    off += ((nf * sizeof(float) + 255) / 256) * 256;
    return p;
  };
  float* deg  = alloc(N);                   // becomes dis in place
  float* ew   = alloc(E);
  float* h    = alloc((size_t)N * HIDF);
  float* bufQ = alloc((size_t)N * HIDF);    // z / q scratch
  float* bufK = alloc((size_t)N * HIDF);
  float* cat  = alloc((size_t)N * CATW);    // [x1 | h | Ah | A2h | A3h]
  float* kvs  = alloc(HIDF * HIDF);
  float* ksum = alloc(HIDF);
  float* hsum = alloc(HIDF);
  float* bsum = alloc(HIDF);
  float* bsq  = alloc(HIDF);
  float* bmean= alloc(HIDF);
  float* brstd= alloc(HIDF);

  const long NH = (long)N * HIDF;
  const int NSL = 50;
  const int per = (((N + NSL - 1) / NSL) + 3) & ~3;

  auto bn_relu = [&](const float* z, const float* g, const float* b, float* out) {
    k_zero<<<1, 128, 0, stream>>>(bsum, HIDF);
    k_zero<<<1, 128, 0, stream>>>(bsq, HIDF);
    k_colsum<<<512, 128, 0, stream>>>(z, HIDF, N, bsum, bsq);
    k_bnfin<<<1, 128, 0, stream>>>(bsum, bsq, bmean, brstd, (float)N);
    k_bnrelu<<<4096, 256, 0, stream>>>(z, bmean, brstd, g, b, out, NH);
  };

  // --- graph normalization ---
  k_zero<<<256, 256, 0, stream>>>(deg, N);
  k_deg<<<2048, 256, 0, stream>>>(ecol, deg, E);
  k_dis<<<(N + 255) / 256, 256, 0, stream>>>(deg, N);
  k_ew<<<2048, 256, 0, stream>>>(erow, ecol, deg, ew, E);

  // --- input projection + BN0 + ReLU ---
  launch_gemm(x, HIDF, W_in, b_in, nullptr, 0, bufQ, HIDF, N, HIDF, HIDF, stream);
  bn_relu(bufQ, bn_g + 0, bn_b + 0, h);

  for (int i = 0; i < 2; ++i) {
    // q, k projections
    launch_gemm(h, HIDF, Wq + (size_t)i * HIDF * HIDF, bq + i * HIDF,
                nullptr, 0, bufQ, HIDF, N, HIDF, HIDF, stream);
    launch_gemm(h, HIDF, Wk + (size_t)i * HIDF * HIDF, bk + i * HIDF,
                nullptr, 0, bufK, HIDF, N, HIDF, HIDF, stream);

    // normalize rows; ksum = kn.sum(0); hsum = h.sum(0)
    k_zero<<<1, 128, 0, stream>>>(ksum, HIDF);
    k_zero<<<1, 128, 0, stream>>>(hsum, HIDF);
    k_rownorm<<<(N + 3) / 4, 128, 0, stream>>>(bufQ, N, nullptr);
    k_rownorm<<<(N + 3) / 4, 128, 0, stream>>>(bufK, N, ksum);
    k_colsum<<<512, 128, 0, stream>>>(h, HIDF, N, hsum, nullptr);

    // kvs = kn^T @ h
    k_zero<<<64, 256, 0, stream>>>(kvs, HIDF * HIDF);
    k_kvs<<<dim3(64, NSL), 32, 0, stream>>>(bufK, h, kvs, N, per);

    // concat buffer: zero, num -> cols 0..127 (bias = hsum), then divide by den
    k_zero<<<8192, 256, 0, stream>>>(cat, (long)N * CATW);
    launch_gemm(bufQ, HIDF, kvs, hsum, nullptr, 0, cat, CATW, N, HIDF, HIDF, stream);
    k_x1div<<<(N + 3) / 4, 128, 0, stream>>>(cat, bufQ, ksum, N);

    // x2: [h | Ah | A2h | A3h] into cols 128..639
    k_copy_h<<<4096, 256, 0, stream>>>(h, cat, NH);
    k_prop<<<(E + 7) / 8, 256, 0, stream>>>(erow, ecol, ew, h, HIDF,
                                            cat + 2 * HIDF, CATW, E);
    k_prop<<<(E + 7) / 8, 256, 0, stream>>>(erow, ecol, ew, cat + 2 * HIDF, CATW,
                                            cat + 3 * HIDF, CATW, E);
    k_prop<<<(E + 7) / 8, 256, 0, stream>>>(erow, ecol, ew, cat + 3 * HIDF, CATW,
                                            cat + 4 * HIDF, CATW, E);

    // z = cat @ Wfc[i] + bfc[i] + h   (residual fused)
    launch_gemm(cat, CATW, Wfc + (size_t)i * CATW * HIDF, bfc + i * HIDF,
                h, HIDF, bufQ, HIDF, N, CATW, HIDF, stream);

    // h = relu(bn_{i+1}(z))
    bn_relu(bufQ, bn_g + (size_t)(i + 1) * HIDF, bn_b + (size_t)(i + 1) * HIDF, h);
  }

  // output projection
  launch_gemm(h, HIDF, W_out, b_out, nullptr, 0, (float*)d_out, OUTC,
              N, HIDF, OUTC, stream);
}